// FullAttention_14963666060008
// MI455X (gfx1250) — compile-verified
//
#include <hip/hip_runtime.h>
#include <math.h>

typedef __attribute__((ext_vector_type(2))) float v2f;
typedef __attribute__((ext_vector_type(8))) float v8f;

#define WAVES_PER_BLOCK 4

// B=2, L=S=2048, H=16, E=D=64, fp32 causal attention (flash-style, online softmax).
// Uses V_WMMA_F32_16X16X4_F32 (full fp32 matrix path) on gfx1250 wave32.
__global__ __launch_bounds__(32 * WAVES_PER_BLOCK, 1)
void fa_causal_f32_kernel(const float* __restrict__ Q,
                          const float* __restrict__ K,
                          const float* __restrict__ V,
                          float* __restrict__ O)
{
    constexpr int B = 2, L = 2048, H = 16, E = 64;
    constexpr int HE = H * E;                 // stride between consecutive sequence rows
    constexpr int NTILES = L / 16;            // 128 q-tiles per head
    constexpr float scale = 0.125f;           // 1/sqrt(64)

    // Per-wave LDS tile for P (16x16), row stride 18 floats:
    //  - 18*r mod 64 distinct for r=0..15  -> conflict-free column reads
    //  - even stride keeps float2 reads 8-byte aligned
    __shared__ float lds[WAVES_PER_BLOCK][16 * 18];

    const int lane = threadIdx.x & 31;
    const int wave = threadIdx.x >> 5;
    const int half = lane >> 4;   // 0/1: selects K-subrange (A/B frag) or M+8 (C/D frag)
    const int ln   = lane & 15;

    // Task mapping: 64 q-tile *pairs* per head; wave handles qt and (127-qt)
    // so total work per wave is uniform (load balance under causal mask).
    const int task   = blockIdx.x * WAVES_PER_BLOCK + wave;   // 0..2047
    const int qpair  = task & 63;
    const int headId = task >> 6;          // b*H + h
    const int b      = headId >> 4;
    const int h      = headId & 15;

    const size_t seqBase = (size_t)b * L * HE + (size_t)h * E;

    #pragma unroll 1
    for (int rep = 0; rep < 2; ++rep) {
        const int qt = (rep == 0) ? qpair : (NTILES - 1 - qpair);
        const int m0 = qt * 16;

        // ---- load Q fragments (A-layout): qa[kk][v] = Q[m0+ln][4kk + 2*half + v]
        v2f qa[16];
        {
            const float* qp = Q + seqBase + (size_t)(m0 + ln) * HE + 2 * half;
            #pragma unroll
            for (int kk = 0; kk < 16; ++kk) {
                qa[kk][0] = qp[kk * 4 + 0];
                qa[kk][1] = qp[kk * 4 + 1];
            }
        }

        v8f acc[4];
        #pragma unroll
        for (int nt = 0; nt < 4; ++nt) acc[nt] = (v8f){0,0,0,0,0,0,0,0};

        float mrow[8], lrow[8];
        #pragma unroll
        for (int g = 0; g < 8; ++g) { mrow[g] = -3.0e38f; lrow[g] = 0.0f; }

        #pragma unroll 1
        for (int jt = 0; jt <= qt; ++jt) {
            // ---- K fragments (B-layout == A-layout for f32 K=4):
            // kb[kk][v] = K[jt*16+ln][4kk + 2*half + v]
            v2f kb[16];
            {
                const float* kp = K + seqBase + (size_t)(jt * 16 + ln) * HE + 2 * half;
                #pragma unroll
                for (int kk = 0; kk < 16; ++kk) {
                    kb[kk][0] = kp[kk * 4 + 0];
                    kb[kk][1] = kp[kk * 4 + 1];
                }
            }

            // ---- S = Q K^T  (16 chained f32 WMMAs cover E=64)
            v8f s = (v8f){0,0,0,0,0,0,0,0};
            #pragma unroll
            for (int kk = 0; kk < 16; ++kk)
                s = __builtin_amdgcn_wmma_f32_16x16x4_f32(
                        false, qa[kk], false, kb[kk], (short)0, s, false, false);

            // ---- scale + causal mask + online softmax (per C/D row g)
            const int ncol = jt * 16 + ln;     // global key index of this lane's column
            #pragma unroll
            for (int g = 0; g < 8; ++g) {
                const int mr = m0 + g + 8 * half;       // global query row
                float sv = s[g] * scale;
                sv = (ncol > mr) ? -3.0e38f : sv;       // branchless causal mask

                // row max over the 16 lanes holding this row (xor masks < 16
                // stay within each wave32 half)
                float tm = sv;
                tm = fmaxf(tm, __shfl_xor(tm, 1, 32));
                tm = fmaxf(tm, __shfl_xor(tm, 2, 32));
                tm = fmaxf(tm, __shfl_xor(tm, 4, 32));
                tm = fmaxf(tm, __shfl_xor(tm, 8, 32));

                const float mnew = fmaxf(mrow[g], tm);
                const float fac  = __expf(mrow[g] - mnew);
                const float p    = __expf(sv - mnew);
                mrow[g] = mnew;

                float ts = p;
                ts += __shfl_xor(ts, 1, 32);
                ts += __shfl_xor(ts, 2, 32);
                ts += __shfl_xor(ts, 4, 32);
                ts += __shfl_xor(ts, 8, 32);
                lrow[g] = lrow[g] * fac + ts;

                #pragma unroll
                for (int nt = 0; nt < 4; ++nt) acc[nt][g] *= fac;

                // stage P into LDS (C-layout position -> [row][col])
                lds[wave][(g + 8 * half) * 18 + ln] = p;
            }

            // ---- reload P as A-fragments: pa[kk][v] = P[ln][4kk + 2*half + v]
            v2f pa[4];
            #pragma unroll
            for (int kk = 0; kk < 4; ++kk) {
                const float* lp = &lds[wave][ln * 18 + kk * 4 + 2 * half];
                pa[kk][0] = lp[0];
                pa[kk][1] = lp[1];
            }

            // ---- O += P V  (K = 16 keys, N = 64 features in 4 tiles)
            const float* vbase = V + seqBase + (size_t)(jt * 16) * HE;
            #pragma unroll
            for (int kk = 0; kk < 4; ++kk) {
                const float* vrow = vbase + (size_t)(kk * 4 + 2 * half) * HE;
                #pragma unroll
                for (int nt = 0; nt < 4; ++nt) {
                    v2f vb;
                    const float* vp = vrow + nt * 16 + ln;
                    vb[0] = vp[0];
                    vb[1] = vp[HE];
                    acc[nt] = __builtin_amdgcn_wmma_f32_16x16x4_f32(
                                  false, pa[kk], false, vb, (short)0, acc[nt],
                                  false, false);
                }
            }
        }

        // ---- epilogue: divide by softmax denominator, store [B,L,H,D]
        #pragma unroll
        for (int g = 0; g < 8; ++g) {
            const float inv = 1.0f / lrow[g];
            const int m = m0 + g + 8 * half;
            float* op = O + seqBase + (size_t)m * HE + ln;
            #pragma unroll
            for (int nt = 0; nt < 4; ++nt)
                op[nt * 16] = acc[nt][g] * inv;
        }
    }
}

extern "C" void kernel_launch(void* const* d_in, const int* in_sizes, int n_in,
                              void* d_out, int out_size, void* d_ws, size_t ws_size,
                              hipStream_t stream) {
    const float* Q = (const float*)d_in[0];
    const float* K = (const float*)d_in[1];
    const float* V = (const float*)d_in[2];
    float* O = (float*)d_out;

    // tasks = B*H*(L/16)/2 pairs = 2048; 4 waves per 128-thread block -> 512 blocks
    constexpr int tasks = 2 * 16 * (2048 / 16) / 2;
    dim3 grid(tasks / WAVES_PER_BLOCK);
    dim3 block(32 * WAVES_PER_BLOCK);
    fa_causal_f32_kernel<<<grid, block, 0, stream>>>(Q, K, V, O);
}